// GAT_37031208026683
// MI455X (gfx1250) — compile-verified
//
#include <hip/hip_runtime.h>
#include <hip/hip_bf16.h>

typedef float v2f __attribute__((ext_vector_type(2)));
typedef float v8f __attribute__((ext_vector_type(8)));

#define NEG_SLOPE 0.2f
#define EPS_F 1e-16f

// ---------------------------------------------------------------------------
// Ordered float <-> uint encoding so segment-max can use integer atomicMax.
// ---------------------------------------------------------------------------
__device__ __forceinline__ unsigned ford(float f) {
    unsigned u = __float_as_uint(f);
    return (u & 0x80000000u) ? ~u : (u | 0x80000000u);
}
__device__ __forceinline__ float funord(unsigned u) {
    u = (u & 0x80000000u) ? (u & 0x7FFFFFFFu) : ~u;
    return __uint_as_float(u);
}

// ---------------------------------------------------------------------------
// Fill kernels (graph-capture safe replacement for hipMemset)
// ---------------------------------------------------------------------------
__global__ void fill_f32(float* p, float v, long long n) {
    long long i = (long long)blockIdx.x * blockDim.x + threadIdx.x;
    if (i < n) p[i] = v;
}
__global__ void fill_u32(unsigned* p, unsigned v, long long n) {
    long long i = (long long)blockIdx.x * blockDim.x + threadIdx.x;
    if (i < n) p[i] = v;
}

// ---------------------------------------------------------------------------
// fp32 WMMA GEMM: C[M,NC] = A[M,K] * B[K,NC], K and NC compile-time.
// One wave32 computes one 16x16 tile via V_WMMA_F32_16X16X4_F32.
//   A 16x4 layout : lane L holds M=L&15, K = {0,1}+2*(L>>4) in v[0],v[1]
//   B 4x16 layout : lane L holds N=L&15, K = {0,1}+2*(L>>4) in v[0],v[1]
//   C 16x16 layout: lane L holds N=L&15, M = r + 8*(L>>4) in v[r], r=0..7
// ---------------------------------------------------------------------------
template <int K, int NC>
__global__ void wmma_gemm(const float* __restrict__ A,
                          const float* __restrict__ B,
                          float* __restrict__ C,
                          int numRowTiles) {
    constexpr int colTiles = NC / 16;
    const int lane   = threadIdx.x & 31;
    const int waveId = (blockIdx.x * blockDim.x + threadIdx.x) >> 5;
    const int rowTile = waveId / colTiles;
    const int colTile = waveId % colTiles;
    if (rowTile >= numRowTiles) return;   // wave-uniform: EXEC stays all-1s

    const int mrow  = lane & 15;
    const int khalf = (lane >> 4) << 1;   // 0 or 2
    const int ncol  = lane & 15;

    const float* __restrict__ Arow = A + (long long)(rowTile * 16 + mrow) * K;
    const float* __restrict__ Bcol = B + colTile * 16 + ncol;

    v8f acc = {0.f, 0.f, 0.f, 0.f, 0.f, 0.f, 0.f, 0.f};
#pragma unroll
    for (int k0 = 0; k0 < K; k0 += 4) {
        v2f a, b;
        a.x = Arow[k0 + khalf];
        a.y = Arow[k0 + khalf + 1];
        b.x = Bcol[(long long)(k0 + khalf) * NC];
        b.y = Bcol[(long long)(k0 + khalf + 1) * NC];
        acc = __builtin_amdgcn_wmma_f32_16x16x4_f32(
            /*neg_a=*/false, a, /*neg_b=*/false, b,
            /*c_mod=*/(short)0, acc, /*reuse_a=*/false, /*reuse_b=*/false);
    }

    float* __restrict__ Crow = C + colTile * 16 + ncol;
    const int mbase = rowTile * 16 + ((lane >> 4) << 3);
#pragma unroll
    for (int r = 0; r < 8; ++r)
        Crow[(long long)(mbase + r) * NC] = acc[r];
}

// ---------------------------------------------------------------------------
// Per-(node, head) attention scores: a_src[n,h] = <h[n,h,:], att_src[h,:]>
// ---------------------------------------------------------------------------
__global__ void attn_scores(const float* __restrict__ hmat,
                            const float* __restrict__ attS,
                            const float* __restrict__ attD,
                            float* __restrict__ asrc, float* __restrict__ adst,
                            int N, int H, int C) {
    long long tid = (long long)blockIdx.x * blockDim.x + threadIdx.x;
    if (tid >= (long long)N * H) return;
    const int h = (int)(tid % H);
    const long long n = tid / H;
    const float* row = hmat + (n * H + h) * (long long)C;
    const float* as  = attS + h * C;
    const float* ad  = attD + h * C;
    float s = 0.f, d = 0.f;
    for (int c = 0; c < C; ++c) {
        const float v = row[c];
        s += v * as[c];
        d += v * ad[c];
    }
    asrc[tid] = s;
    adst[tid] = d;
}

// ---------------------------------------------------------------------------
// Edge kernels. Edge id e: e<E -> (src=ei[e], dst=ei[E+e]); e>=E -> self loop.
// ---------------------------------------------------------------------------
__global__ void edge_score_max(const int* __restrict__ ei, int E, long long EP, int H,
                               const float* __restrict__ asrc,
                               const float* __restrict__ adst,
                               float* __restrict__ ebuf,
                               unsigned* __restrict__ mbuf) {
    long long tid = (long long)blockIdx.x * blockDim.x + threadIdx.x;
    if (tid >= EP * H) return;
    const int h = (int)(tid % H);
    const long long eid = tid / H;
    const int s = (eid < E) ? ei[eid]     : (int)(eid - E);
    const int d = (eid < E) ? ei[E + eid] : (int)(eid - E);
    float v = asrc[(long long)s * H + h] + adst[(long long)d * H + h];
    v = (v > 0.f) ? v : NEG_SLOPE * v;
    ebuf[tid] = v;
    atomicMax(&mbuf[(long long)d * H + h], ford(v));
}

__global__ void edge_expsum(const int* __restrict__ ei, int E, long long EP, int H,
                            float* __restrict__ ebuf,
                            const unsigned* __restrict__ mbuf,
                            float* __restrict__ den) {
    long long tid = (long long)blockIdx.x * blockDim.x + threadIdx.x;
    if (tid >= EP * H) return;
    const int h = (int)(tid % H);
    const long long eid = tid / H;
    const int d = (eid < E) ? ei[E + eid] : (int)(eid - E);
    const float mv = funord(mbuf[(long long)d * H + h]);
    const float ex = expf(ebuf[tid] - mv);
    ebuf[tid] = ex;
    atomicAdd(&den[(long long)d * H + h], ex);
}

__global__ void edge_scatter(const int* __restrict__ ei, int E, long long EP,
                             int H, int C,
                             const float* __restrict__ ebuf,
                             const float* __restrict__ den,
                             const float* __restrict__ hsrc,
                             float* __restrict__ acc) {
    const int Cq = C >> 2;                       // float4 chunks per head
    long long tid = (long long)blockIdx.x * blockDim.x + threadIdx.x;
    const long long tot = EP * H * Cq;
    if (tid >= tot) return;
    const int c4 = (int)(tid % Cq);
    const int h  = (int)((tid / Cq) % H);
    const long long eid = tid / ((long long)H * Cq);
    const int s = (eid < E) ? ei[eid]     : (int)(eid - E);
    const int d = (eid < E) ? ei[E + eid] : (int)(eid - E);
    const float alpha = ebuf[eid * H + h] / (den[(long long)d * H + h] + EPS_F);
    const float4 hv = *(const float4*)(hsrc + ((long long)s * H + h) * C + c4 * 4);
    float* dp = acc + ((long long)d * H + h) * C + c4 * 4;
    atomicAdd(dp + 0, alpha * hv.x);
    atomicAdd(dp + 1, alpha * hv.y);
    atomicAdd(dp + 2, alpha * hv.z);
    atomicAdd(dp + 3, alpha * hv.w);
}

// ---------------------------------------------------------------------------
// Epilogues
// ---------------------------------------------------------------------------
__global__ void bias_relu(const float* __restrict__ acc,
                          const float* __restrict__ b,
                          float* __restrict__ out, long long n, int F) {
    long long i = (long long)blockIdx.x * blockDim.x + threadIdx.x;
    if (i >= n) return;
    const float v = acc[i] + b[(int)(i % F)];
    out[i] = v > 0.f ? v : 0.f;
}

__global__ void bias_add(const float* __restrict__ acc,
                         const float* __restrict__ b,
                         float* __restrict__ out, long long n, int F) {
    long long i = (long long)blockIdx.x * blockDim.x + threadIdx.x;
    if (i >= n) return;
    out[i] = acc[i] + b[(int)(i % F)];
}

// ---------------------------------------------------------------------------
// Host-side launcher
// ---------------------------------------------------------------------------
static inline size_t align256(size_t x) { return (x + 255) & ~(size_t)255; }
static inline int nblk(long long n, int b) { return (int)((n + b - 1) / b); }

extern "C" void kernel_launch(void* const* d_in, const int* in_sizes, int n_in,
                              void* d_out, int out_size, void* d_ws, size_t ws_size,
                              hipStream_t stream) {
    const float* x    = (const float*)d_in[0];
    const int*   ei   = (const int*)d_in[1];
    const float* W1   = (const float*)d_in[2];
    const float* aS1  = (const float*)d_in[3];
    const float* aD1  = (const float*)d_in[4];
    const float* b1   = (const float*)d_in[5];
    const float* W2   = (const float*)d_in[6];
    const float* aS2  = (const float*)d_in[7];
    const float* aD2  = (const float*)d_in[8];
    const float* b2   = (const float*)d_in[9];
    float* out = (float*)d_out;

    const int IN = 128, HC1 = 128, H1 = 4, C1 = 32, H2 = 1, C2 = 32;
    const int N = in_sizes[0] / IN;
    const int E = in_sizes[1] / 2;
    const long long EP = (long long)E + N;

    // workspace layout
    char* ws = (char*)d_ws;
    size_t o = 0;
    float*    h1raw = (float*)(ws + o);    o += align256((size_t)N * HC1 * 4);  // GEMM1 out, then relu'd h1
    float*    acc   = (float*)(ws + o);    o += align256((size_t)N * HC1 * 4);  // layer accumulators
    float*    ebuf  = (float*)(ws + o);    o += align256((size_t)EP * H1 * 4);  // edge scores / exp
    float*    as    = (float*)(ws + o);    o += align256((size_t)N * H1 * 4);
    float*    ad    = (float*)(ws + o);    o += align256((size_t)N * H1 * 4);
    unsigned* mbuf  = (unsigned*)(ws + o); o += align256((size_t)N * H1 * 4);
    float*    den   = (float*)(ws + o);    o += align256((size_t)N * H1 * 4);
    float*    h2raw = (float*)(ws + o);    o += align256((size_t)N * C2 * 4);
    (void)ws_size; (void)n_in; (void)out_size;

    const int B = 256;

    // ========================= Layer 1 (H=4, C=32) =========================
    fill_u32<<<nblk((long long)N * H1, B), B, 0, stream>>>(mbuf, 0u, (long long)N * H1);
    fill_f32<<<nblk((long long)N * H1, B), B, 0, stream>>>(den, 0.f, (long long)N * H1);
    fill_f32<<<nblk((long long)N * HC1, B), B, 0, stream>>>(acc, 0.f, (long long)N * HC1);

    // h = x @ W1  : [N,128] = [N,128]x[128,128]; 8 col-tiles, one wave each
    {
        const int rowTiles = N / 16;                         // N = 100000 -> 6250
        const int waves = rowTiles * (HC1 / 16);
        wmma_gemm<128, 128><<<nblk((long long)waves * 32, B), B, 0, stream>>>(x, W1, h1raw, rowTiles);
    }

    attn_scores<<<nblk((long long)N * H1, B), B, 0, stream>>>(h1raw, aS1, aD1, as, ad, N, H1, C1);
    edge_score_max<<<nblk(EP * H1, B), B, 0, stream>>>(ei, E, EP, H1, as, ad, ebuf, mbuf);
    edge_expsum<<<nblk(EP * H1, B), B, 0, stream>>>(ei, E, EP, H1, ebuf, mbuf, den);
    edge_scatter<<<nblk(EP * H1 * (C1 / 4), B), B, 0, stream>>>(ei, E, EP, H1, C1, ebuf, den, h1raw, acc);

    // h1 = relu(acc + b1)  (overwrites h1raw, which is no longer needed)
    bias_relu<<<nblk((long long)N * HC1, B), B, 0, stream>>>(acc, b1, h1raw, (long long)N * HC1, HC1);

    // ========================= Layer 2 (H=1, C=32) =========================
    fill_u32<<<nblk((long long)N * H2, B), B, 0, stream>>>(mbuf, 0u, (long long)N * H2);
    fill_f32<<<nblk((long long)N * H2, B), B, 0, stream>>>(den, 0.f, (long long)N * H2);
    fill_f32<<<nblk((long long)N * C2, B), B, 0, stream>>>(acc, 0.f, (long long)N * C2);

    // h2 = h1 @ W2 : [N,32] = [N,128]x[128,32]; 2 col-tiles
    {
        const int rowTiles = N / 16;
        const int waves = rowTiles * (C2 / 16);
        wmma_gemm<128, 32><<<nblk((long long)waves * 32, B), B, 0, stream>>>(h1raw, W2, h2raw, rowTiles);
    }

    attn_scores<<<nblk((long long)N * H2, B), B, 0, stream>>>(h2raw, aS2, aD2, as, ad, N, H2, C2);
    edge_score_max<<<nblk(EP * H2, B), B, 0, stream>>>(ei, E, EP, H2, as, ad, ebuf, mbuf);
    edge_expsum<<<nblk(EP * H2, B), B, 0, stream>>>(ei, E, EP, H2, ebuf, mbuf, den);
    edge_scatter<<<nblk(EP * H2 * (C2 / 4), B), B, 0, stream>>>(ei, E, EP, H2, C2, ebuf, den, h2raw, acc);

    // out = acc + b2   (mean over 1 head == identity)
    bias_add<<<nblk((long long)N * C2, B), B, 0, stream>>>(acc, b2, out, (long long)N * C2, C2);
}